// fc_hgnn_58231166599430
// MI455X (gfx1250) — compile-verified
//
#include <hip/hip_runtime.h>
#include <hip/hip_bf16.h>

#define NN    50000
#define EE    1600000
#define FIN   2000
#define DD    20
#define NCOL  160   // 8 matrices * 20 cols, = 10 WMMA tiles exactly

typedef float v2f __attribute__((ext_vector_type(2)));
typedef float v8f __attribute__((ext_vector_type(8)));

// ---------------- weight packing ----------------
__global__ void pack_w0(const float* __restrict__ Wq, const float* __restrict__ Wk,
                        const float* __restrict__ Wv, const float* __restrict__ Ws,
                        const float* __restrict__ bq, const float* __restrict__ bk,
                        const float* __restrict__ bv, const float* __restrict__ bs,
                        float* __restrict__ Wcat, float* __restrict__ bcat) {
    int idx = blockIdx.x * blockDim.x + threadIdx.x;
    if (idx >= FIN * NCOL) return;
    int k = idx / NCOL, c = idx % NCOL;
    int m = c / DD, d = c % DD;
    int t = m >> 2, which = m & 3;
    const float* W = (which == 0) ? Wq : (which == 1) ? Wk : (which == 2) ? Wv : Ws;
    Wcat[idx] = W[((size_t)t * FIN + k) * DD + d];
    if (k == 0) {
        const float* b = (which == 0) ? bq : (which == 1) ? bk : (which == 2) ? bv : bs;
        bcat[c] = b[t * DD + d];
    }
}

__global__ void pack_wl(const float* __restrict__ Wq, const float* __restrict__ Wk,
                        const float* __restrict__ Wv, const float* __restrict__ Ws,
                        const float* __restrict__ bq, const float* __restrict__ bk,
                        const float* __restrict__ bv, const float* __restrict__ bs,
                        int li, float* __restrict__ Wcat, float* __restrict__ bcat) {
    int idx = blockIdx.x * blockDim.x + threadIdx.x;
    if (idx >= DD * NCOL) return;
    int k = idx / NCOL, c = idx % NCOL;
    int m = c / DD, d = c % DD;
    int t = m >> 2, which = m & 3;
    const float* W = (which == 0) ? Wq : (which == 1) ? Wk : (which == 2) ? Wv : Ws;
    Wcat[idx] = W[(((size_t)(li * 2 + t)) * DD + k) * DD + d];
    if (k == 0) {
        const float* b = (which == 0) ? bq : (which == 1) ? bk : (which == 2) ? bv : bs;
        bcat[c] = b[(li * 2 + t) * DD + d];
    }
}

// ---------------- WMMA f32 GEMM: C[M x 160] = A[M x K] * B[K x 160] + bias ----------------
// block = 320 threads = 10 waves; wave w -> N-tile w; blockIdx.x -> M-tile (16 rows)
__global__ __launch_bounds__(320) void gemm_xw(const float* __restrict__ A, int lda, int K,
                                               const float* __restrict__ B,
                                               const float* __restrict__ bias,
                                               float* __restrict__ C) {
    const int lane = threadIdx.x & 31;
    const int wave = threadIdx.x >> 5;
    const int l16  = lane & 15;
    const int hi   = lane >> 4;                  // 0 or 1
    const long mbase = (long)blockIdx.x * 16;
    const float* Arow = A + (mbase + l16) * (long)lda;   // A fragment: M = lane%16
    const int col = wave * 16 + l16;                     // B/C column = N
    const float* Bp = B + col;
    v8f acc = {};
    for (int k0 = 0; k0 < K; k0 += 4) {
        const int ka = k0 + 2 * hi;
        v2f a, b;
        a[0] = Arow[ka];
        a[1] = Arow[ka + 1];
        b[0] = Bp[(long)ka * NCOL];
        b[1] = Bp[(long)(ka + 1) * NCOL];
        acc = __builtin_amdgcn_wmma_f32_16x16x4_f32(false, a, false, b,
                                                    (short)0, acc, false, false);
    }
    const float bv = bias[col];
#pragma unroll
    for (int r = 0; r < 8; ++r) {
        long row = mbase + r + 8 * hi;           // C: M = r + 8*(lane/16)
        C[row * NCOL + col] = acc[r] + bv;
    }
}

// ---------------- edge pass A: logits + segment max (ordered-uint atomicMax) ----------------
__global__ void edge_logits_max(const float* __restrict__ qkvs, const int* __restrict__ eidx,
                                int qoff, int koff,
                                float* __restrict__ logits, unsigned* __restrict__ mkey) {
    int e = blockIdx.x * blockDim.x + threadIdx.x;
    if (e >= EE) return;
    int s = eidx[e];
    int d = eidx[EE + e];
    const float* q = qkvs + (long)d * NCOL + qoff;
    const float* k = qkvs + (long)s * NCOL + koff;
    float acc = 0.f;
#pragma unroll
    for (int i = 0; i < DD; ++i) acc += q[i] * k[i];
    acc *= 0.22360679774997896f;                 // 1/sqrt(20)
    logits[e] = acc;
    unsigned u = __float_as_uint(acc);
    unsigned key = (u & 0x80000000u) ? ~u : (u | 0x80000000u);
    atomicMax(&mkey[d], key);
}

// ---------------- edge pass B: exp, z-sum, unnormalized agg ----------------
__global__ void edge_softmax_agg(const float* __restrict__ qkvs, const int* __restrict__ eidx,
                                 int voff, const float* __restrict__ logits,
                                 const unsigned* __restrict__ mkey,
                                 float* __restrict__ z, float* __restrict__ agg) {
    int e = blockIdx.x * blockDim.x + threadIdx.x;
    if (e >= EE) return;
    int s = eidx[e];
    int d = eidx[EE + e];
    unsigned key = mkey[d];
    unsigned u = (key & 0x80000000u) ? (key & 0x7FFFFFFFu) : ~key;
    float m = __uint_as_float(u);
    float ev = __expf(logits[e] - m);
    atomicAdd(&z[d], ev);
    const float* v = qkvs + (long)s * NCOL + voff;
    float* a = agg + (long)d * DD;
#pragma unroll
    for (int i = 0; i < DD; ++i) atomicAdd(&a[i], ev * v[i]);
}

// ---------------- node combine + BN statistics ----------------
__global__ void node_combine(const float* __restrict__ qkvs,
                             const float* __restrict__ agg_s, const float* __restrict__ z_s,
                             const float* __restrict__ agg_d, const float* __restrict__ z_d,
                             const float* __restrict__ w1, const float* __restrict__ w2, int l,
                             float* __restrict__ xc, float* __restrict__ stats) {
    __shared__ float ssum[DD], ssq[DD];
    if (threadIdx.x < DD) { ssum[threadIdx.x] = 0.f; ssq[threadIdx.x] = 0.f; }
    __syncthreads();
    int n = blockIdx.x * blockDim.x + threadIdx.x;
    if (n < NN) {
        float a = w1[l], b = w2[l];
        float inv = 1.f / (a + b);
        float c1 = a * inv, c2 = b * inv;
        const float* base = qkvs + (long)n * NCOL;
        float izs = 1.f / (z_s[n] + 1e-16f);
        float izd = 1.f / (z_d[n] + 1e-16f);
#pragma unroll
        for (int i = 0; i < DD; ++i) {
            float x1 = agg_s[(long)n * DD + i] * izs + base[60 + i];     // skip (same)
            float x2 = agg_d[(long)n * DD + i] * izd + base[140 + i];    // skip (diff)
            float x = c1 * x1 + c2 * x2;
            xc[(long)n * DD + i] = x;
            atomicAdd(&ssum[i], x);
            atomicAdd(&ssq[i], x * x);
        }
    }
    __syncthreads();
    if (threadIdx.x < DD) {
        atomicAdd(&stats[threadIdx.x], ssum[threadIdx.x]);
        atomicAdd(&stats[DD + threadIdx.x], ssq[threadIdx.x]);
    }
}

// ---------------- BN apply + leaky ReLU ----------------
__global__ void bn_apply(const float* __restrict__ xc, const float* __restrict__ stats,
                         const float* __restrict__ gamma, const float* __restrict__ beta,
                         int l, float* __restrict__ fcbuf, float* __restrict__ xbuf) {
    int idx = blockIdx.x * blockDim.x + threadIdx.x;
    if (idx >= NN * DD) return;
    int i = idx % DD, n = idx / DD;
    float mu  = stats[i] * (1.0f / NN);
    float var = stats[DD + i] * (1.0f / NN) - mu * mu;
    float y = (xc[idx] - mu) * rsqrtf(var + 1e-5f) * gamma[i] + beta[i];
    y = (y > 0.f) ? y : 0.01f * y;
    fcbuf[(long)n * 80 + l * DD + i] = y;
    xbuf[idx] = y;
}

// ---------------- final projection [N,80] @ [80,2] + bias ----------------
__global__ void final_out(const float* __restrict__ fcbuf, const float* __restrict__ Wout,
                          const float* __restrict__ bout, float* __restrict__ out) {
    int n = blockIdx.x * blockDim.x + threadIdx.x;
    if (n >= NN) return;
    float a0 = bout[0], a1 = bout[1];
    const float* f = fcbuf + (long)n * 80;
#pragma unroll
    for (int c = 0; c < 80; ++c) {
        a0 += f[c] * Wout[2 * c];
        a1 += f[c] * Wout[2 * c + 1];
    }
    out[2 * n]     = a0;
    out[2 * n + 1] = a1;
}

extern "C" void kernel_launch(void* const* d_in, const int* in_sizes, int n_in,
                              void* d_out, int out_size, void* d_ws, size_t ws_size,
                              hipStream_t stream) {
    const float* features = (const float*)d_in[0];
    const int*   same_idx = (const int*)d_in[1];
    const int*   diff_idx = (const int*)d_in[2];
    const float* W0q = (const float*)d_in[3];   const float* b0q = (const float*)d_in[4];
    const float* W0k = (const float*)d_in[5];   const float* b0k = (const float*)d_in[6];
    const float* W0v = (const float*)d_in[7];   const float* b0v = (const float*)d_in[8];
    const float* W0s = (const float*)d_in[9];   const float* b0s = (const float*)d_in[10];
    const float* Wq  = (const float*)d_in[11];  const float* bq  = (const float*)d_in[12];
    const float* Wk  = (const float*)d_in[13];  const float* bk  = (const float*)d_in[14];
    const float* Wv  = (const float*)d_in[15];  const float* bv  = (const float*)d_in[16];
    const float* Ws  = (const float*)d_in[17];  const float* bs  = (const float*)d_in[18];
    const float* w1  = (const float*)d_in[19];  const float* w2  = (const float*)d_in[20];
    const float* bn_gamma = (const float*)d_in[21];
    const float* bn_beta  = (const float*)d_in[22];
    const float* Wout = (const float*)d_in[23]; const float* bout = (const float*)d_in[24];
    float* out = (float*)d_out;
    (void)in_sizes; (void)n_in; (void)out_size; (void)ws_size;

    float* ws = (float*)d_ws;
    size_t off = 0;
    auto alloc = [&](size_t nelems) { float* p = ws + off; off += nelems; return p; };
    float*    qkvs    = alloc((size_t)NN * NCOL);    // 32 MB
    float*    Wcat    = alloc((size_t)FIN * NCOL);   // 1.28 MB
    float*    bcat    = alloc(NCOL);
    float*    logit_s = alloc(EE);                   // 6.4 MB
    float*    logit_d = alloc(EE);                   // 6.4 MB
    unsigned* mkey_s  = (unsigned*)alloc(NN);
    unsigned* mkey_d  = (unsigned*)alloc(NN);
    float*    z_s     = alloc(NN);
    float*    z_d     = alloc(NN);
    float*    agg_s   = alloc((size_t)NN * DD);      // 4 MB
    float*    agg_d   = alloc((size_t)NN * DD);      // 4 MB
    float*    xc      = alloc((size_t)NN * DD);      // 4 MB
    float*    xbuf    = alloc((size_t)NN * DD);      // 4 MB
    float*    fcbuf   = alloc((size_t)NN * 80);      // 16 MB
    float*    stats   = alloc(64);

    const int EB = 256;
    const int eblocks = (EE + EB - 1) / EB;
    const int nblocks = (NN + EB - 1) / EB;
    const int dblocks = (NN * DD + EB - 1) / EB;

    for (int l = 0; l < 4; ++l) {
        if (l == 0) {
            pack_w0<<<(FIN * NCOL + 255) / 256, 256, 0, stream>>>(
                W0q, W0k, W0v, W0s, b0q, b0k, b0v, b0s, Wcat, bcat);
            gemm_xw<<<NN / 16, 320, 0, stream>>>(features, FIN, FIN, Wcat, bcat, qkvs);
        } else {
            pack_wl<<<(DD * NCOL + 255) / 256, 256, 0, stream>>>(
                Wq, Wk, Wv, Ws, bq, bk, bv, bs, l - 1, Wcat, bcat);
            gemm_xw<<<NN / 16, 320, 0, stream>>>(xbuf, DD, DD, Wcat, bcat, qkvs);
        }
        hipMemsetAsync(mkey_s, 0, (size_t)NN * 4, stream);
        hipMemsetAsync(mkey_d, 0, (size_t)NN * 4, stream);
        hipMemsetAsync(z_s,    0, (size_t)NN * 4, stream);
        hipMemsetAsync(z_d,    0, (size_t)NN * 4, stream);
        hipMemsetAsync(agg_s,  0, (size_t)NN * DD * 4, stream);
        hipMemsetAsync(agg_d,  0, (size_t)NN * DD * 4, stream);
        hipMemsetAsync(stats,  0, 64 * 4, stream);

        edge_logits_max<<<eblocks, EB, 0, stream>>>(qkvs, same_idx, 0,  20,  logit_s, mkey_s);
        edge_logits_max<<<eblocks, EB, 0, stream>>>(qkvs, diff_idx, 80, 100, logit_d, mkey_d);
        edge_softmax_agg<<<eblocks, EB, 0, stream>>>(qkvs, same_idx, 40,  logit_s, mkey_s, z_s, agg_s);
        edge_softmax_agg<<<eblocks, EB, 0, stream>>>(qkvs, diff_idx, 120, logit_d, mkey_d, z_d, agg_d);

        node_combine<<<nblocks, EB, 0, stream>>>(qkvs, agg_s, z_s, agg_d, z_d, w1, w2, l, xc, stats);
        bn_apply<<<dblocks, EB, 0, stream>>>(xc, stats, bn_gamma + l * DD, bn_beta + l * DD,
                                             l, fcbuf, xbuf);
    }
    final_out<<<nblocks, EB, 0, stream>>>(fcbuf, Wout, bout, out);
}